// GCL_49959059587771
// MI455X (gfx1250) — compile-verified
//
#include <hip/hip_runtime.h>
#include <math.h>

typedef float v2f __attribute__((ext_vector_type(2)));
typedef float v4f __attribute__((ext_vector_type(4)));
typedef float v8f __attribute__((ext_vector_type(8)));

#define NROWS 50000
#define GROWS 512
#define DDIM  256
#define TILES (NROWS / 16)   // 3125, exact

__device__ __forceinline__ float softplus_f(float x) {
    // x in ~[-1,1] here (cosine sims), direct form is accurate
    return log1pf(__expf(x));
}

// ---------------------------------------------------------------------------
// Kernel 1: reciprocal L2 norms of g1/g2 rows. One wave32 per row.
// rg[0..511] = 1/||g1_row||, rg[512..1023] = 1/||g2_row||
// ---------------------------------------------------------------------------
__global__ void gnorm_kernel(const float* __restrict__ g1,
                             const float* __restrict__ g2,
                             float* __restrict__ rg) {
    int wave = (blockIdx.x * blockDim.x + threadIdx.x) >> 5;
    int lane = threadIdx.x & 31;
    if (wave >= 2 * GROWS) return;
    const float* g = (wave < GROWS) ? g1 : g2;
    int row = (wave < GROWS) ? wave : wave - GROWS;

    const float* p = g + (size_t)row * DDIM + lane * 8;   // 32 lanes x 8 = 256
    v4f a = *(const v4f*)(p);
    v4f b = *(const v4f*)(p + 4);
    float ss = a.x*a.x + a.y*a.y + a.z*a.z + a.w*a.w
             + b.x*b.x + b.y*b.y + b.z*b.z + b.w*b.w;
    #pragma unroll
    for (int off = 16; off > 0; off >>= 1)
        ss += __shfl_xor(ss, off, 32);
    if (lane == 0) rg[wave] = 1.0f / fmaxf(sqrtf(ss), 1e-12f);
}

// ---------------------------------------------------------------------------
// Kernel 2: one wave32 per 16-row tile per stream.
//   stream 0: z = z1, b = batch_1, self = g1, cross = g2
//   stream 1: z = z2, b = batch_2, self = g2, cross = g1
// A-matrix (16x4 f32): lane<16 -> row=lane, K={0,1}; lane>=16 -> row=lane-16, K={2,3}
// B-matrix (4x16 f32): lane<16 -> col=lane, K={0,1}; lane>=16 -> col=lane-16, K={2,3}
// Tile dot products = diag of 16x16 f32 accumulator.
// ---------------------------------------------------------------------------
__global__ void pair_jsd_kernel(const float* __restrict__ z1,
                                const float* __restrict__ z2,
                                const float* __restrict__ g1,
                                const float* __restrict__ g2,
                                const int* __restrict__ b1,
                                const int* __restrict__ b2,
                                const float* __restrict__ rg,      // [2*GROWS]
                                float* __restrict__ partial) {     // [2*TILES]
    int wave = (blockIdx.x * blockDim.x + threadIdx.x) >> 5;
    if (wave >= 2 * TILES) return;
    int lane = threadIdx.x & 31;
    int strm = (wave >= TILES) ? 1 : 0;
    int tile = wave - strm * TILES;

    const float* z       = strm ? z2 : z1;
    const int*   batch   = strm ? b2 : b1;
    const float* gSelf   = strm ? g2 : g1;
    const float* gCross  = strm ? g1 : g2;
    const float* rgSelf  = strm ? rg + GROWS : rg;
    const float* rgCross = strm ? rg : rg + GROWS;

    int j  = lane & 15;     // row within tile == N column index
    int kh = lane >> 4;     // K half: offsets {0,1} vs {2,3}
    int row = tile * 16 + j;
    int bsel = batch[row];

    float rgs = rgSelf[bsel];
    float rgc = rgCross[bsel];

    const float* zrow  = z      + (size_t)row  * DDIM + 2 * kh;
    const float* gsrow = gSelf  + (size_t)bsel * DDIM + 2 * kh;
    const float* gcrow = gCross + (size_t)bsel * DDIM + 2 * kh;

    v8f acc_s = {};
    v8f acc_c = {};
    float nacc = 0.0f;      // partial ||z_row||^2 (lane j covers K%4 in {0,1}, lane j+16 covers {2,3})

    #pragma unroll 4
    for (int k0 = 0; k0 < DDIM; k0 += 4) {
        v2f a  = *(const v2f*)(zrow  + k0);
        v2f bs = *(const v2f*)(gsrow + k0);
        v2f bc = *(const v2f*)(gcrow + k0);
        acc_s = __builtin_amdgcn_wmma_f32_16x16x4_f32(
                    false, a, false, bs, (short)0, acc_s, false, false);
        acc_c = __builtin_amdgcn_wmma_f32_16x16x4_f32(
                    false, a, false, bc, (short)0, acc_c, false, false);
        nacc = fmaf(a.x, a.x, nacc);
        nacc = fmaf(a.y, a.y, nacc);
    }

    // Diagonal extraction: D[m][n]: m<8 -> vgpr m, lane n ; m>=8 -> vgpr m-8, lane n+16.
    // So D[r][r] sits on lane r of acc[r]; D[r+8][r+8] on lane r+24 of acc[r].
    float tilesum = 0.0f;
    #pragma unroll
    for (int r = 0; r < 8; ++r) {
        float s0 = __shfl(acc_s[r], r,      32);
        float c0 = __shfl(acc_c[r], r,      32);
        float s1 = __shfl(acc_s[r], r + 24, 32);
        float c1 = __shfl(acc_c[r], r + 24, 32);
        int j0 = r, j1 = r + 8;
        float n0 = __shfl(nacc, j0, 32) + __shfl(nacc, j0 + 16, 32);
        float n1 = __shfl(nacc, j1, 32) + __shfl(nacc, j1 + 16, 32);
        float rz0 = 1.0f / fmaxf(sqrtf(n0), 1e-12f);
        float rz1 = 1.0f / fmaxf(sqrtf(n1), 1e-12f);
        float rgs0 = __shfl(rgs, j0, 32), rgc0 = __shfl(rgc, j0, 32);
        float rgs1 = __shfl(rgs, j1, 32), rgc1 = __shfl(rgc, j1, 32);
        float ss0 = s0 * rz0 * rgs0, cc0 = c0 * rz0 * rgc0;
        float ss1 = s1 * rz1 * rgs1, cc1 = c1 * rz1 * rgc1;
        // ep_jsd(s) - ep_jsd(c) = softplus(-c) - softplus(-s)
        float d0 = softplus_f(-cc0) - softplus_f(-ss0);
        float d1 = softplus_f(-cc1) - softplus_f(-ss1);
        tilesum = fmaf(d0, d0, tilesum);
        tilesum = fmaf(d1, d1, tilesum);
    }
    if (lane == 0) partial[wave] = tilesum;   // wave = strm*TILES + tile
}

// ---------------------------------------------------------------------------
// Kernel 3: deterministic fixed-order reduction -> sqrt(S1)+sqrt(S2)
// ---------------------------------------------------------------------------
__global__ void finalize_kernel(const float* __restrict__ partial,
                                float* __restrict__ out) {
    __shared__ float sh1[256];
    __shared__ float sh2[256];
    int t = threadIdx.x;
    float s1 = 0.0f, s2 = 0.0f;
    for (int i = t; i < TILES; i += 256) {
        s1 += partial[i];
        s2 += partial[TILES + i];
    }
    sh1[t] = s1; sh2[t] = s2;
    __syncthreads();
    #pragma unroll
    for (int off = 128; off > 0; off >>= 1) {
        if (t < off) { sh1[t] += sh1[t + off]; sh2[t] += sh2[t + off]; }
        __syncthreads();
    }
    if (t == 0) out[0] = sqrtf(sh1[0]) + sqrtf(sh2[0]);
}

extern "C" void kernel_launch(void* const* d_in, const int* in_sizes, int n_in,
                              void* d_out, int out_size, void* d_ws, size_t ws_size,
                              hipStream_t stream) {
    const float* z1 = (const float*)d_in[0];
    const float* z2 = (const float*)d_in[1];
    const float* g1 = (const float*)d_in[2];
    const float* g2 = (const float*)d_in[3];
    const int*   b1 = (const int*)d_in[4];
    const int*   b2 = (const int*)d_in[5];
    float* rg      = (float*)d_ws;        // 1024 floats
    float* partial = rg + 2 * GROWS;      // 6250 floats
    float* out     = (float*)d_out;

    // 1024 norm-waves, 8 waves (256 threads) per block
    hipLaunchKernelGGL(gnorm_kernel, dim3((2 * GROWS) / 8), dim3(256), 0, stream,
                       g1, g2, rg);

    // 6250 tile-waves, 8 per block
    int totalWaves = 2 * TILES;
    hipLaunchKernelGGL(pair_jsd_kernel, dim3((totalWaves + 7) / 8), dim3(256), 0, stream,
                       z1, z2, g1, g2, b1, b2, rg, partial);

    hipLaunchKernelGGL(finalize_kernel, dim3(1), dim3(256), 0, stream,
                       partial, out);
}